// DenoiseNet_79087527789092
// MI455X (gfx1250) — compile-verified
//
#include <hip/hip_runtime.h>
#include <hip/hip_bf16.h>

// Problem constants (from the reference)
#define BB 8
#define NN 4096
#define MM 4608
#define NMOD 4
#define QTILES (NN / 16)      // 256 query tiles per (module,batch)
#define TTILES (MM / 16)      // 288 target tiles

typedef float v2f __attribute__((ext_vector_type(2)));
typedef float v8f __attribute__((ext_vector_type(8)));

// ---------------- workspace layout (float4 units) ----------------
// Qaug: [NMOD][B][N]  augmented queries (x,y,z,1)          -> 4*8*4096  float4
// Taug: [3][B][M]     augmented targets (-2x,-2y,-2z,|t|^2) -> 3*8*4608 float4
// acc : 4 floats (per-module loss sums)
#define QAUG_ELEMS (NMOD * BB * NN)
#define TAUG_ELEMS (3 * BB * MM)

__global__ void zero_acc_kernel(float* acc) {
    if (threadIdx.x < 4) acc[threadIdx.x] = 0.0f;
}

// Build augmented queries: pcl_input_0 = noisy - seed ; input_{i+1} = input_i + disp_i
__global__ void prep_queries_kernel(const float* __restrict__ noisy,
                                    const float* __restrict__ seeds,
                                    const float* __restrict__ disp,
                                    float4* __restrict__ qaug) {
    int idx = blockIdx.x * blockDim.x + threadIdx.x;
    if (idx >= BB * NN) return;
    int b = idx >> 12;          // / N
    float sx = seeds[b * 3 + 0];
    float sy = seeds[b * 3 + 1];
    float sz = seeds[b * 3 + 2];
    float qx = noisy[idx * 3 + 0] - sx;
    float qy = noisy[idx * 3 + 1] - sy;
    float qz = noisy[idx * 3 + 2] - sz;
#pragma unroll
    for (int i = 0; i < NMOD; ++i) {
        qaug[i * (BB * NN) + idx] = make_float4(qx, qy, qz, 1.0f);
        if (i < NMOD - 1) {
            const float* dp = disp + (size_t)(i * (BB * NN) + idx) * 3;
            qx += dp[0];
            qy += dp[1];
            qz += dp[2];
        }
    }
}

// Build augmented targets: set0 = clean-seed + noise0*std/4,
// set1 = clean-seed + noise1*std/16, set2 = clean-seed.
__global__ void prep_targets_kernel(const float* __restrict__ clean,
                                    const float* __restrict__ seeds,
                                    const float* __restrict__ stdv,
                                    const float* __restrict__ noise,
                                    float4* __restrict__ taug) {
    int idx = blockIdx.x * blockDim.x + threadIdx.x;
    if (idx >= BB * MM) return;
    int b = idx / MM;
    float sx = seeds[b * 3 + 0];
    float sy = seeds[b * 3 + 1];
    float sz = seeds[b * 3 + 2];
    float cx = clean[idx * 3 + 0] - sx;
    float cy = clean[idx * 3 + 1] - sy;
    float cz = clean[idx * 3 + 2] - sz;
    float s  = stdv[b];
    float s0 = s * 0.25f;
    float s1 = s * 0.0625f;
    const float* n0 = noise + (size_t)idx * 3;
    const float* n1 = noise + (size_t)(BB * MM + idx) * 3;

    float tx, ty, tz, tt;
    // set 0
    tx = cx + n0[0] * s0; ty = cy + n0[1] * s0; tz = cz + n0[2] * s0;
    tt = tx * tx + ty * ty + tz * tz;
    taug[0 * (BB * MM) + idx] = make_float4(-2.0f * tx, -2.0f * ty, -2.0f * tz, tt);
    // set 1
    tx = cx + n1[0] * s1; ty = cy + n1[1] * s1; tz = cz + n1[2] * s1;
    tt = tx * tx + ty * ty + tz * tz;
    taug[1 * (BB * MM) + idx] = make_float4(-2.0f * tx, -2.0f * ty, -2.0f * tz, tt);
    // set 2 (clean)
    tt = cx * cx + cy * cy + cz * cz;
    taug[2 * (BB * MM) + idx] = make_float4(-2.0f * cx, -2.0f * cy, -2.0f * cz, tt);
}

// Main WMMA nearest-neighbor + loss kernel.
// One wave (32 lanes) handles 16 queries vs all 4608 targets.
// d2(m,n) = A(m,:) * B(:,n) with A row = (qx,qy,qz,1), B col = (-2tx,-2ty,-2tz,|t|^2)
// computed by V_WMMA_F32_16X16X4_F32 (K=4 exact fit).
__global__ void __launch_bounds__(256)
nn_loss_kernel(const float4* __restrict__ qaug,
               const float4* __restrict__ taug,
               const float* __restrict__ disp,
               float* __restrict__ acc) {
    const int lane = threadIdx.x & 31;
    const int wave = threadIdx.x >> 5;
    const int gt   = blockIdx.x * (blockDim.x >> 5) + wave;  // 0..8191
    const int mod  = gt >> 11;                               // 2048 tiles per module
    const int rem  = gt & 2047;
    const int b    = rem >> 8;
    const int qt   = rem & 255;
    const int qbase = qt << 4;
    const int tset = (mod < 2) ? mod : 2;
    const int lh   = lane & 15;

    const float4* qptr = qaug + ((mod * BB + b) << 12);      // * N
    const float4* tptr = taug + (tset * BB + b) * MM;

    // A operand: 16x4 f32 layout — lanes 0-15: K=0,1 ; lanes 16-31: K=2,3
    float4 qa = qptr[qbase + lh];
    v2f A;
    A[0] = (lane < 16) ? qa.x : qa.z;
    A[1] = (lane < 16) ? qa.y : 1.0f;  // augmented 4th component = 1

    float mval[8];
    int   midx[8];
#pragma unroll
    for (int v = 0; v < 8; ++v) { mval[v] = 3.4e38f; midx[v] = 0; }

    for (int j = 0; j < TTILES; ++j) {
        float4 tb = tptr[(j << 4) + lh];
        v2f Bm;
        Bm[0] = (lane < 16) ? tb.x : tb.z;
        Bm[1] = (lane < 16) ? tb.y : tb.w;
        v8f c = {};
        v8f d = __builtin_amdgcn_wmma_f32_16x16x4_f32(
            false, A, false, Bm, (short)0, c, false, false);
        const int tindex = (j << 4) + lh;
#pragma unroll
        for (int v = 0; v < 8; ++v) {
            bool lt = d[v] < mval[v];
            mval[v] = lt ? d[v] : mval[v];
            midx[v] = lt ? tindex : midx[v];
        }
    }

    // Butterfly min over the 16 target columns (stays within each 16-lane half).
#pragma unroll
    for (int off = 8; off >= 1; off >>= 1) {
#pragma unroll
        for (int v = 0; v < 8; ++v) {
            float ov = __shfl_xor(mval[v], off, 32);
            int   oi = __shfl_xor(midx[v], off, 32);
            bool take = (ov < mval[v]) || (ov == mval[v] && oi < midx[v]);
            mval[v] = take ? ov : mval[v];
            midx[v] = take ? oi : midx[v];
        }
    }

    // Lane 0 owns rows 0..7 (VGPR v -> M=v); lane 16 owns rows 8..15 (M=v+8).
    if (lh == 0) {
        const int rowoff = (lane >> 4) << 3;  // 0 or 8
        float sum = 0.0f;
        for (int v = 0; v < 8; ++v) {
            int qrow = qbase + rowoff + v;
            float4 q = qptr[qrow];
            float4 t = tptr[midx[v]];
            float nx = -0.5f * t.x;  // recover target point from -2*t storage
            float ny = -0.5f * t.y;
            float nz = -0.5f * t.z;
            const float* dp = disp + (size_t)(((mod * BB + b) << 12) + qrow) * 3;
            float ex = dp[0] - (nx - q.x);
            float ey = dp[1] - (ny - q.y);
            float ez = dp[2] - (nz - q.z);
            sum += ex * ex + ey * ey + ez * ez;
        }
        atomicAdd(&acc[mod], sum);
    }
}

__global__ void finalize_kernel(const float* __restrict__ acc, float* __restrict__ out) {
    if (blockIdx.x == 0 && threadIdx.x == 0) {
        // loss_i = (sum over b,n) / B ; final = sum_i loss_i
        float t = (acc[0] + acc[1] + acc[2] + acc[3]) * (1.0f / (float)BB);
        out[0] = t;
        out[1] = t;
    }
}

extern "C" void kernel_launch(void* const* d_in, const int* in_sizes, int n_in,
                              void* d_out, int out_size, void* d_ws, size_t ws_size,
                              hipStream_t stream) {
    (void)in_sizes; (void)n_in; (void)out_size; (void)ws_size;

    const float* noisy = (const float*)d_in[0];  // (B,N,3)
    const float* clean = (const float*)d_in[1];  // (B,M,3)
    const float* seeds = (const float*)d_in[2];  // (B,1,3)
    const float* stdv  = (const float*)d_in[3];  // (B,)
    const float* disp  = (const float*)d_in[4];  // (4,B,N,3)
    const float* noise = (const float*)d_in[5];  // (2,B,M,3)
    float* out = (float*)d_out;

    float4* qaug = (float4*)d_ws;
    float4* taug = qaug + QAUG_ELEMS;
    float*  acc  = (float*)(taug + TAUG_ELEMS);

    zero_acc_kernel<<<1, 32, 0, stream>>>(acc);

    {
        int total = BB * NN;
        prep_queries_kernel<<<(total + 255) / 256, 256, 0, stream>>>(noisy, seeds, disp, qaug);
    }
    {
        int total = BB * MM;
        prep_targets_kernel<<<(total + 255) / 256, 256, 0, stream>>>(clean, seeds, stdv, noise, taug);
    }
    {
        // 4 modules * 8 batches * 256 query-tiles = 8192 waves ; 8 waves/block
        nn_loss_kernel<<<NMOD * BB * QTILES / 8, 256, 0, stream>>>(qaug, taug, disp, acc);
    }
    finalize_kernel<<<1, 32, 0, stream>>>(acc, out);
}